// RateRNN_34668976013831
// MI455X (gfx1250) — compile-verified
//
#include <hip/hip_runtime.h>

// ---------------------------------------------------------------------------
// RateRNN on MI455X (gfx1250).
//
// Per timestep (sequential chain, T=256):
//   rec[64,2048] = r[64,2048] @ W_rec^T[2048,2048]      (f32 WMMA 16x16x4)
//   du = (-u + f + rec + bias + eta) * tau_inv
//   u  = clip(u + du*DT, -5, 5);  r = relu(u)  (precomputed for next step)
// One kernel launch per step (graph-replayed); u and r ping-pong in d_ws.
// W_rec (16 MB) stays resident in the 192 MB L2 across all 256 steps.
// K-chunks are staged to LDS with GLOBAL_LOAD_ASYNC_TO_LDS_B128 (ASYNCcnt)
// and double-buffered so the next chunk's fetch overlaps the WMMA work.
// ---------------------------------------------------------------------------

typedef float v2f __attribute__((ext_vector_type(2)));
typedef float v8f __attribute__((ext_vector_type(8)));
typedef int   v4i __attribute__((ext_vector_type(4)));

static constexpr int NH     = 2048;
static constexpr int MROWS  = 64;        // B*TR
static constexpr int TSTEPS = 256;
static constexpr int NEXC   = 1638;
static constexpr int KC     = 64;        // K chunk staged in LDS
static constexpr int LDP    = KC + 4;    // pad: bank = (4*row+k)%64 -> conflict-free
static constexpr int NCHUNK = NH / KC;   // 32
static constexpr int CHTILE = MROWS * LDP; // floats per staged tile (64*68)

#if defined(__has_builtin)
#if __has_builtin(__builtin_amdgcn_global_load_async_to_lds_b128)
#define HAS_ASYNC_LDS 1
#endif
#endif
#ifndef HAS_ASYNC_LDS
#define HAS_ASYNC_LDS 0
#endif

#if HAS_ASYNC_LDS
#if __has_builtin(__builtin_amdgcn_s_wait_asynccnt)
#define WAIT_ASYNCCNT(n) __builtin_amdgcn_s_wait_asynccnt(n)
#else
#define WAIT_ASYNCCNT(n) asm volatile("s_wait_asynccnt %0" ::"i"(n) : "memory")
#endif

// Builtin parameter types (from hipcc diagnostic): global-AS pointer to v4i,
// LDS-AS pointer to v4i, immediate offset, immediate cpol.
typedef __attribute__((address_space(1))) v4i gv4i;
typedef __attribute__((address_space(3))) v4i sv4i;

__device__ __forceinline__ void async_copy16(const float* g, float* l)
{
    // copies 16 bytes from global address g to LDS address l, tracked by ASYNCcnt (per-lane addresses)
    __builtin_amdgcn_global_load_async_to_lds_b128((gv4i*)g, (sv4i*)l, 0, 0);
}
#endif

// WMMA body for one staged K-chunk: 16 K-iters x 4 M-tiles of
// v_wmma_f32_16x16x4_f32. Fragment layouts per ISA 7.12.2:
//   A 16x4 : lanes0-15 (K=0/1 in the 2 VGPRs), lanes16-31 (K=2/3)
//   B 4x16 : mirrored; C/D: VGPR v -> M = v + 8*(lane>=16), N = lane&15
__device__ __forceinline__ void mma_chunk(const float* __restrict__ As,
                                          const float* __restrict__ Bs,
                                          int lo, int kOff, int wave, v8f acc[4])
{
#pragma unroll
    for (int k0 = 0; k0 < KC; k0 += 4) {
        v2f b  = *(const v2f*)(Bs + (wave * 16 + lo) * LDP + k0 + kOff);
        v2f a0 = *(const v2f*)(As + ( 0 + lo) * LDP + k0 + kOff);
        v2f a1 = *(const v2f*)(As + (16 + lo) * LDP + k0 + kOff);
        v2f a2 = *(const v2f*)(As + (32 + lo) * LDP + k0 + kOff);
        v2f a3 = *(const v2f*)(As + (48 + lo) * LDP + k0 + kOff);
        acc[0] = __builtin_amdgcn_wmma_f32_16x16x4_f32(false, a0, false, b, (short)0, acc[0], false, false);
        acc[1] = __builtin_amdgcn_wmma_f32_16x16x4_f32(false, a1, false, b, (short)0, acc[1], false, false);
        acc[2] = __builtin_amdgcn_wmma_f32_16x16x4_f32(false, a2, false, b, (short)0, acc[2], false, false);
        acc[3] = __builtin_amdgcn_wmma_f32_16x16x4_f32(false, a3, false, b, (short)0, acc[3], false, false);
    }
}

__global__ __launch_bounds__(128)
void rnn_step_kernel(const float* __restrict__ u_in,    // [64,2048] state
                     const float* __restrict__ r_in,    // [64,2048] relu(state)
                     float*       __restrict__ u_out,
                     float*       __restrict__ r_out,
                     const float* __restrict__ fs,      // [64,256,2048]
                     const float* __restrict__ noise,   // [64,257,2048]
                     const float* __restrict__ W,       // [2048,2048]
                     const float* __restrict__ bias,    // [2048]
                     float*       __restrict__ u_hist,  // [64,256,2048]
                     float*       __restrict__ du_hist, // [64,256,2048]
                     int t)
{
    const int tid   = threadIdx.x;
    const int wave  = tid >> 5;          // 0..3 : owns N sub-tile wave*16
    const int lane  = tid & 31;
    const int hf    = lane >> 4;
    const int lo    = lane & 15;
    const int kOff  = hf << 1;
    const int nBase = blockIdx.x * 64;   // 32 blocks cover N=2048

    // Double-buffered staging: As = r chunk [64xKC], Bs[j][k] = W[nBase+j][kBase+k]
    __shared__ __align__(16) float As[2][CHTILE];   // 2 x 17.4 KB
    __shared__ __align__(16) float Bs[2][CHTILE];   // 2 x 17.4 KB

    v8f acc[4] = {};

    // Per-thread staging slice: 8 float4 of As + 8 float4 of Bs per chunk.
    const int sRow = tid >> 4;            // 0..7 base row
    const int sC4  = (tid & 15) << 2;     // float4 column offset

#if HAS_ASYNC_LDS
    // ---- prologue: async-issue chunk 0 into buffer 0 ----
    {
#pragma unroll
        for (int j = 0; j < 8; ++j) {
            const int row = sRow + j * 8;
            async_copy16(r_in + (size_t)row * NH + sC4, &As[0][row * LDP + sC4]);
        }
#pragma unroll
        for (int j = 0; j < 8; ++j) {
            const int row = sRow + j * 8;
            async_copy16(W + (size_t)(nBase + row) * NH + sC4, &Bs[0][row * LDP + sC4]);
        }
    }
    for (int kc = 0; kc < NCHUNK; ++kc) {
        const int cur = kc & 1;
        if (kc + 1 < NCHUNK) {
            const int nxt   = cur ^ 1;
            const int kNext = (kc + 1) * KC;
#pragma unroll
            for (int j = 0; j < 8; ++j) {
                const int row = sRow + j * 8;
                async_copy16(r_in + (size_t)row * NH + kNext + sC4, &As[nxt][row * LDP + sC4]);
            }
#pragma unroll
            for (int j = 0; j < 8; ++j) {
                const int row = sRow + j * 8;
                async_copy16(W + (size_t)(nBase + row) * NH + kNext + sC4, &Bs[nxt][row * LDP + sC4]);
            }
            WAIT_ASYNCCNT(16);   // 16 next-chunk ops in flight; current chunk done
        } else {
            WAIT_ASYNCCNT(0);
        }
        __syncthreads();                       // current chunk visible to all waves
        mma_chunk(As[cur], Bs[cur], lo, kOff, wave, acc);
        __syncthreads();                       // safe to overwrite buffer next iter
    }
#else
    // ---- fallback: synchronous staging, single buffer ----
    for (int kc = 0; kc < NCHUNK; ++kc) {
        const int kBase = kc * KC;
#pragma unroll
        for (int j = 0; j < 8; ++j) {
            const int row = sRow + j * 8;
            *(float4*)(&As[0][row * LDP + sC4]) =
                *(const float4*)(r_in + (size_t)row * NH + kBase + sC4);
            *(float4*)(&Bs[0][row * LDP + sC4]) =
                *(const float4*)(W + (size_t)(nBase + row) * NH + kBase + sC4);
        }
        __syncthreads();
        mma_chunk(As[0], Bs[0], lo, kOff, wave, acc);
        __syncthreads();
    }
#endif

    // ---- epilogue: elementwise RNN update on the 64x16 tile this wave owns ----
    const int   col  = nBase + (wave << 4) + lo;
    const float bcol = bias[col];
    const float tinv = (col < NEXC) ? 50.0f : 100.0f;   // 1/0.02 : 1/0.01

#pragma unroll
    for (int m = 0; m < 4; ++m) {
#pragma unroll
        for (int v = 0; v < 8; ++v) {
            const int    R    = (m << 4) + (hf << 3) + v;
            const size_t sidx = (size_t)R * NH + col;
            const size_t hidx = ((size_t)R * TSTEPS + t) * NH + col;
            const float  uv   = u_in[sidx];
            const float  f    = fs[hidx];
            const float  eta  = noise[((size_t)R * (TSTEPS + 1) + t) * NH + col];
            const float  rec  = acc[m][v];
            const float  du   = (-uv + f + rec + bcol + eta) * tinv;
            float        un   = uv + du * 0.001f;
            un = fminf(5.0f, fmaxf(-5.0f, un));
            u_out[sidx]   = un;
            r_out[sidx]   = fmaxf(un, 0.0f);   // relu state for next step's GEMM
            u_hist[hidx]  = un;
            du_hist[hidx] = du;
        }
    }
}

__global__ __launch_bounds__(256)
void init_state_kernel(const float* __restrict__ u0,
                       float* __restrict__ u_buf, float* __restrict__ r_buf)
{
    const size_t i = (size_t)blockIdx.x * blockDim.x + threadIdx.x;
    if (i < (size_t)MROWS * NH) {
        const float u = u0[i];
        u_buf[i] = u;
        r_buf[i] = fmaxf(u, 0.0f);
    }
}

__global__ __launch_bounds__(256)
void eta_final_kernel(const float* __restrict__ noise, float* __restrict__ out)
{
    const size_t i = (size_t)blockIdx.x * blockDim.x + threadIdx.x;
    if (i < (size_t)MROWS * NH) {
        const size_t r = i >> 11, n = i & (NH - 1);
        out[i] = noise[(r * (TSTEPS + 1) + TSTEPS) * NH + n];
    }
}

extern "C" void kernel_launch(void* const* d_in, const int* in_sizes, int n_in,
                              void* d_out, int out_size, void* d_ws, size_t ws_size,
                              hipStream_t stream)
{
    const float* u0    = (const float*)d_in[0];
    const float* fs    = (const float*)d_in[1];
    const float* noise = (const float*)d_in[2];
    const float* W     = (const float*)d_in[3];
    const float* bias  = (const float*)d_in[4];

    const size_t HIST  = (size_t)MROWS * TSTEPS * NH;   // 33,554,432
    const size_t STATE = (size_t)MROWS * NH;            // 131,072
    float* u_hist  = (float*)d_out;
    float* eta_f   = u_hist + HIST;
    float* du_hist = eta_f + STATE;

    // Workspace: u ping-pong + r(=relu(u)) ping-pong (4 x 512 KB).
    float* u_bufs[2] = { (float*)d_ws,            (float*)d_ws + STATE };
    float* r_bufs[2] = { (float*)d_ws + 2 * STATE, (float*)d_ws + 3 * STATE };

    init_state_kernel<<<dim3((MROWS * NH + 255) / 256), dim3(256), 0, stream>>>(
        u0, u_bufs[0], r_bufs[0]);

    for (int t = 0; t < TSTEPS; ++t) {
        const int in = t & 1, outb = in ^ 1;
        rnn_step_kernel<<<dim3(32), dim3(128), 0, stream>>>(
            u_bufs[in], r_bufs[in], u_bufs[outb], r_bufs[outb],
            fs, noise, W, bias, u_hist, du_hist, t);
    }
    eta_final_kernel<<<dim3((MROWS * NH + 255) / 256), dim3(256), 0, stream>>>(noise, eta_f);
}